// GNN_73332271612551
// MI455X (gfx1250) — compile-verified
//
#include <hip/hip_runtime.h>
#include <math.h>

typedef float v2f __attribute__((ext_vector_type(2)));
typedef float v8f __attribute__((ext_vector_type(8)));

// ---------------- degree / normalization ----------------

__global__ void k_init_deg(float* __restrict__ deg, int n) {
  int i = blockIdx.x * blockDim.x + threadIdx.x;
  if (i < n) deg[i] = 1.0f;              // self-loop contributes 1
}

__global__ void k_accum_deg(const int* __restrict__ row, float* __restrict__ deg, int e) {
  int i = blockIdx.x * blockDim.x + threadIdx.x;
  if (i < e) atomicAdd(&deg[row[i]], 1.0f);
}

__global__ void k_rsqrt_inplace(float* __restrict__ deg, int n) {
  int i = blockIdx.x * blockDim.x + threadIdx.x;
  if (i < n) {
    float d = deg[i];
    deg[i] = (d > 0.0f) ? rsqrtf(d) : 0.0f;
  }
}

// ---------------- layer-1 GEMM via WMMA f32 16x16x4 ----------------
// One wave32 per 16-row tile.  hs1[i][:] = dinv[i] * (x @ W1)[i][:]
// t1 initialized to hs1 (the self-loop term of the aggregation).
__global__ void k_gemm1_wmma(const float* __restrict__ x,     // [n,17]
                             const float* __restrict__ W1,    // [17,16]
                             const float* __restrict__ dinv,  // [n]
                             float* __restrict__ hs1,         // [n,16]
                             float* __restrict__ t1,          // [n,16]
                             int n) {
  int wave  = (blockIdx.x * blockDim.x + threadIdx.x) >> 5;
  int lane  = threadIdx.x & 31;
  int row0  = wave << 4;
  if (row0 >= n) return;                 // wave-uniform: EXEC stays all-1 for WMMA

  int m     = lane & 15;                 // A-matrix row / B,D column (per ISA layout)
  int khalf = lane >> 4;                 // which K pair this lane holds

  int arow = row0 + m;
  if (arow >= n) arow = n - 1;           // clamp (tail safety; OOB rows never stored)
  const float* xrow = x + (size_t)arow * 17;

  v8f acc = {0.f, 0.f, 0.f, 0.f, 0.f, 0.f, 0.f, 0.f};

  // k-tiles 0..3 cover K = 0..15: all loads in-range for every lane, no predication.
#pragma unroll
  for (int kt = 0; kt < 4; ++kt) {
    int kb = kt * 4 + khalf * 2;
    v2f a, b;
    a.x = xrow[kb];                      // A: 16x4, lane=M, VGPR0/1 = K pair
    a.y = xrow[kb + 1];
    b.x = W1[kb * 16 + m];               // B: 4x16, lane=N, VGPR0/1 = K pair
    b.y = W1[(kb + 1) * 16 + m];
    acc = __builtin_amdgcn_wmma_f32_16x16x4_f32(
        false, a, false, b, (short)0, acc, false, false);
  }

  // K tail (K = 16): load unconditionally (valid for all lanes: x rows have 17
  // elements, W1 has 272 >= 256+15), zero the upper lane-half via select —
  // branch-free, EXEC stays all-1s.
  {
    float xa = xrow[16];
    float wa = W1[256 + m];
    v2f a, b;
    a.x = khalf ? 0.f : xa;  a.y = 0.f;
    b.x = khalf ? 0.f : wa;  b.y = 0.f;
    acc = __builtin_amdgcn_wmma_f32_16x16x4_f32(
        false, a, false, b, (short)0, acc, false, false);
  }

  // D layout: VGPR v -> row (row0 + 8*khalf + v), col = m. Each lane's 8 rows
  // are contiguous, so dinv loads are contiguous per lane.
  int rbase = row0 + (khalf << 3);
  if (row0 + 16 <= n) {
    // fast path: full tile, branch-free epilogue
    const float* dv = dinv + rbase;
    float* hp = hs1 + (size_t)rbase * 16 + m;
    float* tp = t1  + (size_t)rbase * 16 + m;
#pragma unroll
    for (int v = 0; v < 8; ++v) {
      float val = acc[v] * dv[v];
      hp[(size_t)v * 16] = val;
      tp[(size_t)v * 16] = val;
    }
  } else {
    // tail tile (unused when n % 16 == 0, kept for generality)
#pragma unroll
    for (int v = 0; v < 8; ++v) {
      int r = rbase + v;
      if (r < n) {
        float val = acc[v] * dinv[r];
        size_t idx = (size_t)r * 16 + m;
        hs1[idx] = val;
        t1[idx]  = val;
      }
    }
  }
}

// ---------------- edge scatter, 16 features: one (edge,feature) per lane ----------------
__global__ void k_scatter16(const int* __restrict__ row, const int* __restrict__ col,
                            const float* __restrict__ hs1, float* __restrict__ t1,
                            int e) {
  long long tid = (long long)blockIdx.x * blockDim.x + threadIdx.x;
  int       f   = (int)(tid & 15);
  long long ei  = tid >> 4;
  if (ei < (long long)e) {
    int r = row[ei];
    int c = col[ei];
    atomicAdd(&t1[(size_t)r * 16 + f], hs1[(size_t)c * 16 + f]);
  }
}

// ---------------- layer-1 epilogue fused with layer-2 linear ----------------
// h1 = relu(dinv*t1 + b1);  hs2 = dinv * (h1 @ W2);  t2 initialized to hs2 (self-loop).
__global__ void k_l1post_l2pre(const float* __restrict__ t1,   // [n,16]
                               const float* __restrict__ dinv, // [n]
                               const float* __restrict__ b1,   // [16]
                               const float* __restrict__ W2,   // [16,2]
                               float* __restrict__ hs2,        // [n,2]
                               float* __restrict__ t2,         // [n,2]
                               int n) {
  int i = blockIdx.x * blockDim.x + threadIdx.x;
  if (i >= n) return;
  float di = dinv[i];
  const float* t = t1 + (size_t)i * 16;
  float z0 = 0.f, z1 = 0.f;
#pragma unroll
  for (int f = 0; f < 16; ++f) {
    float h = fmaxf(di * t[f] + b1[f], 0.f);
    z0 = fmaf(h, W2[f * 2 + 0], z0);
    z1 = fmaf(h, W2[f * 2 + 1], z1);
  }
  z0 *= di;
  z1 *= di;
  hs2[(size_t)i * 2 + 0] = z0;
  hs2[(size_t)i * 2 + 1] = z1;
  t2[(size_t)i * 2 + 0]  = z0;
  t2[(size_t)i * 2 + 1]  = z1;
}

// ---------------- edge scatter, 2 features ----------------
__global__ void k_scatter2(const int* __restrict__ row, const int* __restrict__ col,
                           const float* __restrict__ hs2, float* __restrict__ t2,
                           int e) {
  long long tid = (long long)blockIdx.x * blockDim.x + threadIdx.x;
  int       f   = (int)(tid & 1);
  long long ei  = tid >> 1;
  if (ei < (long long)e) {
    int r = row[ei];
    int c = col[ei];
    atomicAdd(&t2[(size_t)r * 2 + f], hs2[(size_t)c * 2 + f]);
  }
}

// ---------------- layer-2 epilogue: bias + log_softmax(2) ----------------
__global__ void k_finalize(const float* __restrict__ t2, const float* __restrict__ dinv,
                           const float* __restrict__ b2, float* __restrict__ out, int n) {
  int i = blockIdx.x * blockDim.x + threadIdx.x;
  if (i >= n) return;
  float di = dinv[i];
  float z0 = di * t2[(size_t)i * 2 + 0] + b2[0];
  float z1 = di * t2[(size_t)i * 2 + 1] + b2[1];
  float mx  = fmaxf(z0, z1);
  float lse = mx + logf(expf(z0 - mx) + expf(z1 - mx));
  out[(size_t)i * 2 + 0] = z0 - lse;
  out[(size_t)i * 2 + 1] = z1 - lse;
}

extern "C" void kernel_launch(void* const* d_in, const int* in_sizes, int n_in,
                              void* d_out, int out_size, void* d_ws, size_t ws_size,
                              hipStream_t stream) {
  const float* x  = (const float*)d_in[0];   // [n,17]
  const int*   ei = (const int*)d_in[1];     // [2,e] int32
  const float* W1 = (const float*)d_in[2];   // [17,16]
  const float* b1 = (const float*)d_in[3];   // [16]
  const float* W2 = (const float*)d_in[4];   // [16,2]
  const float* b2 = (const float*)d_in[5];   // [2]

  int n = in_sizes[0] / 17;
  int e = in_sizes[1] / 2;
  const int* row = ei;
  const int* col = ei + e;

  // workspace layout (floats): dinv[n] | hs1[16n] | t1[16n] | hs2[2n] | t2[2n]  (~44.4 MB)
  float* ws   = (float*)d_ws;
  float* dinv = ws;
  float* hs1  = ws + (size_t)n;
  float* t1   = ws + (size_t)n * 17;
  float* hs2  = ws + (size_t)n * 33;
  float* t2   = ws + (size_t)n * 35;

  const int B  = 256;
  int gn = (n + B - 1) / B;
  int ge = (e + B - 1) / B;

  k_init_deg<<<gn, B, 0, stream>>>(dinv, n);
  k_accum_deg<<<ge, B, 0, stream>>>(row, dinv, e);
  k_rsqrt_inplace<<<gn, B, 0, stream>>>(dinv, n);

  int nwaves = (n + 15) / 16;                         // one wave32 per 16-row tile
  int gw = (nwaves * 32 + B - 1) / B;
  k_gemm1_wmma<<<gw, B, 0, stream>>>(x, W1, dinv, hs1, t1, n);

  long long w16 = (long long)e * 16;
  k_scatter16<<<(int)((w16 + B - 1) / B), B, 0, stream>>>(row, col, hs1, t1, e);

  k_l1post_l2pre<<<gn, B, 0, stream>>>(t1, dinv, b1, W2, hs2, t2, n);

  long long w2 = (long long)e * 2;
  k_scatter2<<<(int)((w2 + B - 1) / B), B, 0, stream>>>(row, col, hs2, t2, e);

  k_finalize<<<gn, B, 0, stream>>>(t2, dinv, b2, (float*)d_out, n);
}